// PatchRandomApplier_78202764525672
// MI455X (gfx1250) — compile-verified
//
#include <hip/hip_runtime.h>
#include <hip/hip_bf16.h>

// Problem constants from the reference setup_inputs():
//   img:   (B=8, C=3, H=1024, W=1024) f32
//   patch: (C=3, P=300, P=300)        f32
//   boxes: (B=8, N=16, 4)             i32  (x1, y1, x2, y2)
// MAX_PATCH_RATIO = 0.9
#define PB_B 8
#define PB_C 3
#define PB_H 1024
#define PB_W 1024
#define PB_N 16
#define PB_P 300

// Tile: blockDim (32, 8); each thread owns 4 consecutive x pixels (float4).
// Tile footprint: 128 x 8 pixels.
#define TILE_W 128
#define TILE_H 8

__device__ __forceinline__ void set_lane(float4& v, int j, float x) {
  // j is a constant under #pragma unroll -> folds to a plain component write.
  if (j == 0) v.x = x; else if (j == 1) v.y = x; else if (j == 2) v.z = x; else v.w = x;
}

__global__ __launch_bounds__(256) void PatchApply_kernel(
    const float* __restrict__ img,
    const float* __restrict__ patch,
    const int*   __restrict__ boxes,
    float*       __restrict__ out)
{
  // Disable FMA contraction: the JAX reference evaluates muls/adds separately.
  #pragma clang fp contract(off)

  const int qx = blockIdx.x * TILE_W + threadIdx.x * 4;  // quad base x
  const int y  = blockIdx.y * TILE_H + threadIdx.y;
  const int b  = blockIdx.z;
  if (qx >= PB_W || y >= PB_H) return;

  const size_t hw  = (size_t)PB_H * PB_W;
  const size_t pix = (size_t)y * PB_W + qx;
  const float* ip = img + (size_t)b * PB_C * hw + pix;
  float*       op = out + (size_t)b * PB_C * hw + pix;

  const int4* bb = (const int4*)(boxes + (size_t)b * PB_N * 4);
  const int tx0 = blockIdx.x * TILE_W;   // uniform tile bounds
  const int ty0 = blockIdx.y * TILE_H;

  // ---- Box scan (single merged loop): last covering box wins -------------
  // Geometry is uniform (s_load + SALU) and consumed in-iteration, so the
  // only persistent per-thread state is last[4] + the winner's 4 params.
  int last[4] = {-1, -1, -1, -1};
  int wys[4], wxs[4], wth[4], wtw[4];

  for (int n = 0; n < PB_N; ++n) {
    const int4 bx = bb[n];                 // uniform -> s_load_b128
    const int bw = bx.z - bx.x;
    const int bh = bx.w - bx.y;
    // th = max(1, int(bh * 0.9f))  (f32 multiply, truncate toward zero)
    int th = (int)((float)bh * 0.9f); th = th < 1 ? 1 : th;
    int tw = (int)((float)bw * 0.9f); tw = tw < 1 ? 1 : tw;
    const int ys = bx.y + (bh - th) / 2;   // operands >= 0 -> '/' == floordiv
    const int xs = bx.x + (bw - tw) / 2;

    // Uniform tile-vs-box rejection: SALU compare + scalar branch skips the
    // per-lane tests for boxes that miss this 128x8 tile (~avg <1 box hits).
    const bool hit = (ys < ty0 + TILE_H) & (ty0 < ys + th) &
                     (xs < tx0 + TILE_W) & (tx0 < xs + tw);
    if (hit) {
      const bool rowIn = (y >= ys) & (y < ys + th);
      #pragma unroll
      for (int j = 0; j < 4; ++j) {
        const int xj = qx + j;
        const bool in = rowIn & (xj >= xs) & (xj < xs + tw);
        if (in) { last[j] = n; wys[j] = ys; wxs[j] = xs; wth[j] = th; wtw[j] = tw; }
      }
    }
  }

  // ---- Pass-through loads: only if some subpixel is uncovered ------------
  // Fully covered quads are write-only (the big bandwidth win: no img read).
  const bool anyUncovered =
      (last[0] < 0) | (last[1] < 0) | (last[2] < 0) | (last[3] < 0);

  float4 c0 = make_float4(0.f, 0.f, 0.f, 0.f), c1 = c0, c2 = c0;
  if (anyUncovered) {
    // gfx1250 prefetch: same lines the B128 loads fetch -> zero wasted HBM
    // bandwidth, just gets the lines moving before the load queue.
    __builtin_prefetch(ip,          0, 0);   // global_prefetch_b8
    __builtin_prefetch(ip + hw,     0, 0);
    __builtin_prefetch(ip + 2 * hw, 0, 0);
    c0 = *(const float4*)(ip);               // global_load_b128 x3
    c1 = *(const float4*)(ip + hw);
    c2 = *(const float4*)(ip + 2 * hw);
  }

  // ---- Bilinear patch samples for covered subpixels ----------------------
  #pragma unroll
  for (int j = 0; j < 4; ++j) {
    if (last[j] >= 0) {
      const float thp = (float)wth[j];
      const float twp = (float)wtw[j];
      const float rr = (float)(y - wys[j]);        // exact (ints < 2^24)
      const float cc = (float)(qx + j - wxs[j]);

      float sy = (rr + 0.5f) * ((float)PB_P / thp) - 0.5f;
      float sx = (cc + 0.5f) * ((float)PB_P / twp) - 0.5f;
      sy = fmaxf(sy, 0.0f);
      sx = fmaxf(sx, 0.0f);

      const int y0 = (int)sy;                      // sy >= 0 -> trunc == floor
      const int x0 = (int)sx;
      const float ly = sy - (float)y0;
      const float lx = sx - (float)x0;

      const int y0c = y0     < PB_P - 1 ? y0     : PB_P - 1;  // y0 >= 0 already
      const int y1c = y0 + 1 < PB_P - 1 ? y0 + 1 : PB_P - 1;
      const int x0c = x0     < PB_P - 1 ? x0     : PB_P - 1;
      const int x1c = x0 + 1 < PB_P - 1 ? x0 + 1 : PB_P - 1;

      const float w00 = (1.0f - ly) * (1.0f - lx);
      const float w01 = (1.0f - ly) * lx;
      const float w10 = ly * (1.0f - lx);
      const float w11 = ly * lx;

      const int i00 = y0c * PB_P + x0c;
      const int i01 = y0c * PB_P + x1c;
      const int i10 = y1c * PB_P + x0c;
      const int i11 = y1c * PB_P + x1c;

      #pragma unroll
      for (int c = 0; c < PB_C; ++c) {
        const float* pc = patch + c * (PB_P * PB_P);         // L2-resident
        // clip(patch, 0, 1) applied at load (reference clips before gather).
        const float p00 = fminf(fmaxf(pc[i00], 0.0f), 1.0f);
        const float p01 = fminf(fmaxf(pc[i01], 0.0f), 1.0f);
        const float p10 = fminf(fmaxf(pc[i10], 0.0f), 1.0f);
        const float p11 = fminf(fmaxf(pc[i11], 0.0f), 1.0f);
        const float v = w00 * p00 + w01 * p01 + w10 * p10 + w11 * p11;
        if (c == 0)      set_lane(c0, j, v);
        else if (c == 1) set_lane(c1, j, v);
        else             set_lane(c2, j, v);
      }
    }
  }

  *(float4*)(op)          = c0;    // global_store_b128 x3
  *(float4*)(op + hw)     = c1;
  *(float4*)(op + 2 * hw) = c2;
}

extern "C" void kernel_launch(void* const* d_in, const int* in_sizes, int n_in,
                              void* d_out, int out_size, void* d_ws, size_t ws_size,
                              hipStream_t stream) {
  const float* img   = (const float*)d_in[0];
  const float* patch = (const float*)d_in[1];
  const int*   boxes = (const int*)d_in[2];
  float*       out   = (float*)d_out;
  (void)in_sizes; (void)n_in; (void)out_size; (void)d_ws; (void)ws_size;

  dim3 block(32, 8, 1);                           // 8 wave32s per block
  dim3 grid(PB_W / TILE_W, PB_H / TILE_H, PB_B);  // 8 x 128 x 8 tiles
  PatchApply_kernel<<<grid, block, 0, stream>>>(img, patch, boxes, out);
}